// DGCNN_16303695856093
// MI455X (gfx1250) — compile-verified
//
#include <hip/hip_runtime.h>

#define N_PTS 4096
#define KNN   41
#define SPAD  48      // spatial tile pad: 3 x 16
#define EPSF  1e-5f
#define NEG_INF -3.402823466e38f

typedef __attribute__((ext_vector_type(2))) float v2f;
typedef __attribute__((ext_vector_type(8))) float v8f;

// ---------------------------------------------------------------------------
// fp32 WMMA: D(16x16) = A(16x4) * B(4x16) + C   (V_WMMA_F32_16X16X4_F32)
// ---------------------------------------------------------------------------
__device__ __forceinline__ v8f wmma_f32_16x16x4(v2f a, v2f b, v8f c) {
  return __builtin_amdgcn_wmma_f32_16x16x4_f32(
      /*neg_a=*/false, a, /*neg_b=*/false, b,
      /*c_mod=*/(short)0, c, /*reuse_a=*/false, /*reuse_b=*/false);
}

// ---------------------------------------------------------------------------
// Kernel A: per-(b,n) KNN.  pd = -||x_i - x_j||^2, take indices of 41 largest
// (tie-break: lower index first, matching jax.lax.top_k).
// One 256-thread block per row; distances live in LDS.
// ---------------------------------------------------------------------------
__global__ __launch_bounds__(256) void knn_kernel(const float* __restrict__ x,
                                                  int* __restrict__ idxOut) {
  __shared__ float dist[N_PTS];
  __shared__ float wval[8];
  __shared__ int   widx[8];
  const int tid  = threadIdx.x;
  const int bid  = blockIdx.x;               // b * N_PTS + i
  const int b    = bid / N_PTS;
  const int i    = bid - b * N_PTS;
  const float* xb = x + (size_t)b * 3 * N_PTS;
  const float xi0 = xb[0 * N_PTS + i];
  const float xi1 = xb[1 * N_PTS + i];
  const float xi2 = xb[2 * N_PTS + i];

  for (int j = tid; j < N_PTS; j += 256) {
    const float d0 = xb[0 * N_PTS + j] - xi0;
    const float d1 = xb[1 * N_PTS + j] - xi1;
    const float d2 = xb[2 * N_PTS + j] - xi2;
    dist[j] = -(d0 * d0 + d1 * d1 + d2 * d2);
  }
  __syncthreads();

  const int lane = tid & 31;
  const int wave = tid >> 5;
  for (int it = 0; it < KNN; ++it) {
    float bv = NEG_INF;
    int   bi = N_PTS;                        // sentinel: larger than any index
    for (int j = tid; j < N_PTS; j += 256) { // ascending j -> '>' keeps low idx
      const float v = dist[j];
      if (v > bv) { bv = v; bi = j; }
    }
    // wave32 reduction (max value, then min index)
#pragma unroll
    for (int off = 16; off > 0; off >>= 1) {
      const float ov = __shfl_xor(bv, off, 32);
      const int   oi = __shfl_xor(bi, off, 32);
      if (ov > bv || (ov == bv && oi < bi)) { bv = ov; bi = oi; }
    }
    if (lane == 0) { wval[wave] = bv; widx[wave] = bi; }
    __syncthreads();
    if (tid == 0) {
      float fv = wval[0];
      int   fi = widx[0];
#pragma unroll
      for (int w = 1; w < 8; ++w) {
        if (wval[w] > fv || (wval[w] == fv && widx[w] < fi)) {
          fv = wval[w]; fi = widx[w];
        }
      }
      idxOut[(size_t)bid * KNN + it] = fi;
      dist[fi] = NEG_INF;
    }
    __syncthreads();
  }
}

// ---------------------------------------------------------------------------
// One conv+BN+ReLU layer on an LDS-resident [Cin x 48] -> [Cout x 48] tile.
// Each wave owns a 16(out-ch) x 16(spatial) WMMA tile; K-loop in steps of 4.
// A-matrix lane layout: lane l holds row (l&15), K = {0,1} (lanes 0-15) or
// {2,3} (lanes 16-31).  B-matrix: lane l holds col (l&15), same K split.
// D: VGPR r = row r (lanes 0-15) / r+8 (lanes 16-31).
// ---------------------------------------------------------------------------
__device__ __forceinline__ void conv_layer(
    const float* __restrict__ w,  const float* __restrict__ gw,
    const float* __restrict__ bw, const float* __restrict__ mw,
    const float* __restrict__ vw,
    int Cin, int CinPad, int Cout,
    const float* __restrict__ in, float* __restrict__ out, int tid) {
  const int lane  = tid & 31;
  const int wave  = tid >> 5;
  const int tiles = (Cout >> 4) * 3;         // (Cout/16) x (48/16)
  const int m15   = lane & 15;
  const int kbase = (lane >> 4) << 1;        // 0 or 2
  const int ohalf = (lane >> 4) << 3;        // 0 or 8

  for (int t = wave; t < tiles; t += 8) {
    const int otile = t / 3;
    const int stile = t - otile * 3;
    const int mrow  = otile * 16 + m15;
    const int col   = stile * 16 + m15;
    v8f acc = {0.f, 0.f, 0.f, 0.f, 0.f, 0.f, 0.f, 0.f};
    for (int kk = 0; kk < CinPad; kk += 4) {
      const int c0 = kk + kbase;
      v2f a, bv;
      a.x  = (c0     < Cin) ? w[mrow * Cin + c0    ] : 0.f;
      a.y  = (c0 + 1 < Cin) ? w[mrow * Cin + c0 + 1] : 0.f;
      bv.x = in[(c0    ) * SPAD + col];
      bv.y = in[(c0 + 1) * SPAD + col];
      acc = wmma_f32_16x16x4(a, bv, acc);
    }
#pragma unroll
    for (int r = 0; r < 8; ++r) {
      const int   o  = otile * 16 + ohalf + r;
      const float sc = gw[o] * __frsqrt_rn(vw[o] + EPSF);
      const float sh = bw[o] - mw[o] * sc;
      const float y  = acc[r] * sc + sh;
      out[o * SPAD + col] = fmaxf(y, 0.f);
    }
  }
}

__device__ __forceinline__ void pool_max(const float* __restrict__ buf,
                                         int Cout, int catOff, int b, int n,
                                         float* __restrict__ cat, int tid) {
  for (int c = tid; c < Cout; c += 256) {
    const float* row = buf + c * SPAD;
    float mx = row[0];
#pragma unroll 8
    for (int k = 1; k < KNN; ++k) mx = fmaxf(mx, row[k]);
    cat[((size_t)b * 512 + catOff + c) * N_PTS + n] = mx;
  }
}

// ---------------------------------------------------------------------------
// Kernel B: fully fused edge-conv per point n.
//   build feat(6x41) in LDS (+ write feat output), then
//   conv1..conv4 + BN + ReLU entirely in LDS, max-pool each into cat[b,512,N].
// Dynamic LDS: two 256x48 fp32 ping-pong buffers (96 KB).
// ---------------------------------------------------------------------------
__global__ __launch_bounds__(256) void edgeconv_kernel(
    const float* __restrict__ x, const int* __restrict__ idx,
    const float* __restrict__ w1, const float* __restrict__ g1,
    const float* __restrict__ b1, const float* __restrict__ m1,
    const float* __restrict__ v1,
    const float* __restrict__ w2, const float* __restrict__ g2,
    const float* __restrict__ b2, const float* __restrict__ m2,
    const float* __restrict__ v2,
    const float* __restrict__ w3, const float* __restrict__ g3,
    const float* __restrict__ b3, const float* __restrict__ m3,
    const float* __restrict__ v3,
    const float* __restrict__ w4, const float* __restrict__ g4,
    const float* __restrict__ b4, const float* __restrict__ m4,
    const float* __restrict__ v4,
    float* __restrict__ featOut, float* __restrict__ cat) {
  extern __shared__ float smem[];
  float* s0 = smem;                 // 256 x 48
  float* s1 = smem + 256 * SPAD;    // 256 x 48

  const int tid = threadIdx.x;
  const int bid = blockIdx.x;       // b * N_PTS + n
  const int b   = bid / N_PTS;
  const int n   = bid - b * N_PTS;
  const float* xb = x + (size_t)b * 3 * N_PTS;
  const float c0v = xb[0 * N_PTS + n];
  const float c1v = xb[1 * N_PTS + n];
  const float c2v = xb[2 * N_PTS + n];

  // zero padding: rows 6,7 fully; cols 41..47 of rows 0..5
  if (tid < SPAD) {
    s0[6 * SPAD + tid] = 0.f;
    s0[7 * SPAD + tid] = 0.f;
    if (tid >= KNN) {
#pragma unroll
      for (int c = 0; c < 6; ++c) s0[c * SPAD + tid] = 0.f;
    }
  }
  if (tid < KNN) {
    const int   j  = idx[(size_t)bid * KNN + tid];
    const float n0 = xb[0 * N_PTS + j];
    const float n1 = xb[1 * N_PTS + j];
    const float n2 = xb[2 * N_PTS + j];
    s0[0 * SPAD + tid] = n0;
    s0[1 * SPAD + tid] = n1;
    s0[2 * SPAD + tid] = n2;
    s0[3 * SPAD + tid] = c0v;
    s0[4 * SPAD + tid] = c1v;
    s0[5 * SPAD + tid] = c2v;
    // feat output: [b, 6, N, K]
    const size_t fbase = (((size_t)b * 6) * N_PTS + n) * KNN + tid;
    const size_t fstep = (size_t)N_PTS * KNN;
    featOut[fbase + 0 * fstep] = n0;
    featOut[fbase + 1 * fstep] = n1;
    featOut[fbase + 2 * fstep] = n2;
    featOut[fbase + 3 * fstep] = c0v;
    featOut[fbase + 4 * fstep] = c1v;
    featOut[fbase + 5 * fstep] = c2v;
  }
  __syncthreads();

  conv_layer(w1, g1, b1, m1, v1,   6,   8,  64, s0, s1, tid);
  __syncthreads();
  pool_max(s1,  64,   0, b, n, cat, tid);
  __syncthreads();

  conv_layer(w2, g2, b2, m2, v2,  64,  64,  64, s1, s0, tid);
  __syncthreads();
  pool_max(s0,  64,  64, b, n, cat, tid);
  __syncthreads();

  conv_layer(w3, g3, b3, m3, v3,  64,  64, 128, s0, s1, tid);
  __syncthreads();
  pool_max(s1, 128, 128, b, n, cat, tid);
  __syncthreads();

  conv_layer(w4, g4, b4, m4, v4, 128, 128, 256, s1, s0, tid);
  __syncthreads();
  pool_max(s0, 256, 256, b, n, cat, tid);
}

// ---------------------------------------------------------------------------
// Kernel C: conv5 — out[b,256,N] = relu(BN(W5(256x512) @ cat[b,512,N])).
// One wave per 16x16 tile; K-loop over 512 channels in steps of 4.
// ---------------------------------------------------------------------------
__global__ __launch_bounds__(32) void conv5_kernel(
    const float* __restrict__ cat, const float* __restrict__ w5,
    const float* __restrict__ g5, const float* __restrict__ b5,
    const float* __restrict__ m5, const float* __restrict__ v5,
    float* __restrict__ out) {
  const int lane  = threadIdx.x;
  const int ntile = blockIdx.x;   // N / 16
  const int otile = blockIdx.y;   // 256 / 16
  const int b     = blockIdx.z;
  const int m15   = lane & 15;
  const int kbase = (lane >> 4) << 1;
  const int ohalf = (lane >> 4) << 3;
  const int mrow  = otile * 16 + m15;
  const int col   = ntile * 16 + m15;
  const float* catB = cat + (size_t)b * 512 * N_PTS;

  v8f acc = {0.f, 0.f, 0.f, 0.f, 0.f, 0.f, 0.f, 0.f};
  for (int c = 0; c < 512; c += 4) {
    const int c0 = c + kbase;
    if (c + 16 < 512) {
      __builtin_prefetch(&catB[(size_t)(c0 + 16) * N_PTS + col], 0, 1);
    }
    v2f a, bv;
    a.x  = w5[mrow * 512 + c0];
    a.y  = w5[mrow * 512 + c0 + 1];
    bv.x = catB[(size_t)(c0    ) * N_PTS + col];
    bv.y = catB[(size_t)(c0 + 1) * N_PTS + col];
    acc = wmma_f32_16x16x4(a, bv, acc);
  }
#pragma unroll
  for (int r = 0; r < 8; ++r) {
    const int   o  = otile * 16 + ohalf + r;
    const float sc = g5[o] * __frsqrt_rn(v5[o] + EPSF);
    const float sh = b5[o] - m5[o] * sc;
    const float y  = acc[r] * sc + sh;
    out[((size_t)b * 256 + o) * N_PTS + col] = fmaxf(y, 0.f);
  }
}

// ---------------------------------------------------------------------------
extern "C" void kernel_launch(void* const* d_in, const int* in_sizes, int n_in,
                              void* d_out, int out_size, void* d_ws, size_t ws_size,
                              hipStream_t stream) {
  (void)in_sizes; (void)n_in; (void)out_size; (void)ws_size;
  const float* x = (const float*)d_in[0];
  const float *W[5], *G[5], *Bb[5], *M[5], *V[5];
  for (int i = 0; i < 5; ++i) {
    W[i]  = (const float*)d_in[1 + i * 5 + 0];
    G[i]  = (const float*)d_in[1 + i * 5 + 1];
    Bb[i] = (const float*)d_in[1 + i * 5 + 2];
    M[i]  = (const float*)d_in[1 + i * 5 + 3];
    V[i]  = (const float*)d_in[1 + i * 5 + 4];
  }
  float* out  = (float*)d_out;                       // (2,256,4096)
  float* feat = out + (size_t)2 * 256 * N_PTS;       // (2,6,4096,41)

  int*   idx = (int*)d_ws;
  size_t idxBytes = (((size_t)2 * N_PTS * KNN * sizeof(int)) + 255u) & ~(size_t)255u;
  float* cat = (float*)((char*)d_ws + idxBytes);     // (2,512,4096)

  knn_kernel<<<2 * N_PTS, 256, 0, stream>>>(x, idx);

  const size_t ldsBytes = (size_t)2 * 256 * SPAD * sizeof(float);  // 96 KB
  edgeconv_kernel<<<2 * N_PTS, 256, ldsBytes, stream>>>(
      x, idx,
      W[0], G[0], Bb[0], M[0], V[0],
      W[1], G[1], Bb[1], M[1], V[1],
      W[2], G[2], Bb[2], M[2], V[2],
      W[3], G[3], Bb[3], M[3], V[3],
      feat, cat);

  conv5_kernel<<<dim3(N_PTS / 16, 256 / 16, 2), 32, 0, stream>>>(
      cat, W[4], G[4], Bb[4], M[4], V[4], out);
}